// HyperGroup_81217831567432
// MI455X (gfx1250) — compile-verified
//
#include <hip/hip_runtime.h>
#include <hip/hip_bf16.h>
#include <math.h>

#define MAXM 20
#define DIM  64

typedef __attribute__((ext_vector_type(2))) float v2f;
typedef __attribute__((ext_vector_type(8))) float v8f;

// ---------------------------------------------------------------------------
// zero a float buffer (graph-capture-safe replacement for memset)
// ---------------------------------------------------------------------------
__global__ void zero_f32(float* __restrict__ p, int n) {
    int i = blockIdx.x * blockDim.x + threadIdx.x;
    if (i < n) p[i] = 0.0f;
}

// ---------------------------------------------------------------------------
// Per-group init: masked-mean embedding (raw membership incl. duplicates),
// distinct-member count m_cnt[g], and deg[u] += 1 per distinct member.
// One block (64 threads = 2 waves) per group; thread t owns dim t.
// ---------------------------------------------------------------------------
__global__ void group_init_k(const int* __restrict__ membership,
                             const float* __restrict__ member_mask,
                             const float* __restrict__ user_emb,
                             float* __restrict__ emb,
                             float* __restrict__ m_cnt,
                             float* __restrict__ deg) {
    int g = blockIdx.x;
    int t = threadIdx.x;
    __shared__ int   mem[MAXM];
    __shared__ float w[MAXM];
    __shared__ int   uniq[MAXM];
    if (t < MAXM) {
        mem[t] = membership[g * MAXM + t];
        w[t]   = expf(member_mask[g * MAXM + t]);
    }
    __syncthreads();
    if (t < MAXM) {
        int u = 1;
        for (int j = 0; j < t; ++j)
            if (mem[j] == mem[t]) { u = 0; break; }
        uniq[t] = u;
    }
    __syncthreads();
    float wsum = 0.0f;
    #pragma unroll
    for (int j = 0; j < MAXM; ++j) wsum += w[j];
    float s = 0.0f;
    for (int j = 0; j < MAXM; ++j)
        s += w[j] * user_emb[(size_t)mem[j] * DIM + t];
    emb[(size_t)g * DIM + t] = s / wsum;
    if (t == 0) {
        int c = 0;
        #pragma unroll
        for (int j = 0; j < MAXM; ++j) c += uniq[j];
        m_cnt[g] = (float)c;
    }
    if (t < MAXM && uniq[t]) atomicAdd(&deg[mem[t]], 1.0f);
}

// ---------------------------------------------------------------------------
// neigh_member[g] = sum over distinct members u of (deg[u]-1) * user_emb[u]
// (the closed form of (H*deg - H) @ U for binary H)
// ---------------------------------------------------------------------------
__global__ void neigh_member_k(const int* __restrict__ membership,
                               const float* __restrict__ deg,
                               const float* __restrict__ user_emb,
                               float* __restrict__ neigh_member) {
    int g = blockIdx.x;
    int t = threadIdx.x;
    __shared__ int mem[MAXM];
    __shared__ int uniq[MAXM];
    if (t < MAXM) mem[t] = membership[g * MAXM + t];
    __syncthreads();
    if (t < MAXM) {
        int u = 1;
        for (int j = 0; j < t; ++j)
            if (mem[j] == mem[t]) { u = 0; break; }
        uniq[t] = u;
    }
    __syncthreads();
    float s = 0.0f;
    for (int j = 0; j < MAXM; ++j)
        if (uniq[j])
            s += (deg[mem[j]] - 1.0f) * user_emb[(size_t)mem[j] * DIM + t];
    neigh_member[(size_t)g * DIM + t] = s;
}

// ---------------------------------------------------------------------------
// H^T @ emb as a scatter: user_acc[u] += emb[g] for each distinct member u
// ---------------------------------------------------------------------------
__global__ void scatter_k(const int* __restrict__ membership,
                          const float* __restrict__ emb,
                          float* __restrict__ user_acc) {
    int g = blockIdx.x;
    int t = threadIdx.x;
    __shared__ int mem[MAXM];
    __shared__ int uniq[MAXM];
    if (t < MAXM) mem[t] = membership[g * MAXM + t];
    __syncthreads();
    if (t < MAXM) {
        int u = 1;
        for (int j = 0; j < t; ++j)
            if (mem[j] == mem[t]) { u = 0; break; }
        uniq[t] = u;
    }
    __syncthreads();
    float e = emb[(size_t)g * DIM + t];
    for (int j = 0; j < MAXM; ++j)
        if (uniq[j]) atomicAdd(&user_acc[(size_t)mem[j] * DIM + t], e);
}

// ---------------------------------------------------------------------------
// msg = H(H^T emb) - m_cnt*emb + neigh_member ; emb_cat = [emb | msg]  [G,128]
// ---------------------------------------------------------------------------
__global__ void gather_cat_k(const int* __restrict__ membership,
                             const float* __restrict__ user_acc,
                             const float* __restrict__ emb,
                             const float* __restrict__ neigh_member,
                             const float* __restrict__ m_cnt,
                             float* __restrict__ emb_cat) {
    int g = blockIdx.x;
    int t = threadIdx.x;
    __shared__ int mem[MAXM];
    __shared__ int uniq[MAXM];
    if (t < MAXM) mem[t] = membership[g * MAXM + t];
    __syncthreads();
    if (t < MAXM) {
        int u = 1;
        for (int j = 0; j < t; ++j)
            if (mem[j] == mem[t]) { u = 0; break; }
        uniq[t] = u;
    }
    __syncthreads();
    float s = 0.0f;
    for (int j = 0; j < MAXM; ++j)
        if (uniq[j]) s += user_acc[(size_t)mem[j] * DIM + t];
    float e   = emb[(size_t)g * DIM + t];
    float msg = s - m_cnt[g] * e + neigh_member[(size_t)g * DIM + t];
    emb_cat[(size_t)g * 2 * DIM + t]       = e;
    emb_cat[(size_t)g * 2 * DIM + DIM + t] = msg;
}

// ---------------------------------------------------------------------------
// GCN transform via fp32 WMMA:  emb_new = l2norm( emb_cat @ Wk[it]^T + bk[it] )
// Block = 128 threads (4 waves), covers a 16-row x 64-col output tile.
// Wave w computes the 16x16 N-tile at n0 = 16*w, K = 128 -> 32x
// v_wmma_f32_16x16x4_f32 per wave. EXEC is all-1s (no divergence here),
// as the WMMA ISA requires. A-tile staged in LDS with +4 float row padding
// to avoid 16-way bank conflicts on the column-strided A-fragment reads.
// ---------------------------------------------------------------------------
__global__ void __launch_bounds__(128)
gcn_transform_wmma(const float* __restrict__ emb_cat,
                   const float* __restrict__ Wk,   // [K][64][128]
                   const float* __restrict__ bk,   // [K][64]
                   float* __restrict__ emb,        // [G][64]  (output)
                   int iter) {
    __shared__ float A[16][132];     // 16 x 128 tile of emb_cat (+pad)
    __shared__ float C[16][DIM];     // 16 x 64 output tile
    __shared__ float nrm[16];

    const int g0 = blockIdx.x * 16;
    const int t  = threadIdx.x;

    // stage A tile: 2048 floats / 128 threads
    for (int i = t; i < 16 * 128; i += 128)
        A[i >> 7][i & 127] = emb_cat[(size_t)g0 * 128 + i];
    __syncthreads();

    const int lane = t & 31;
    const int wave = t >> 5;
    const int n0   = wave * 16;
    const int m    = lane & 15;          // row (A/C) or col (B/C) within tile
    const int hi   = (lane >> 4) & 1;    // lane half
    const int koff = hi * 2;             // ISA: hi lanes hold K+2,K+3

    const float* W = Wk + (size_t)iter * DIM * 2 * DIM + (size_t)(n0 + m) * 2 * DIM;

    v8f acc = {0.f, 0.f, 0.f, 0.f, 0.f, 0.f, 0.f, 0.f};
    #pragma unroll
    for (int k = 0; k < 128; k += 4) {
        v2f a, b;
        // A 16x4 fragment: lane<16 -> (M=lane, K=k..k+1), lane>=16 -> K=k+2..k+3
        a.x = A[m][k + koff];
        a.y = A[m][k + koff + 1];
        // B 4x16 fragment: VGPR r -> K = k + r (+2 for hi half), N = m
        b.x = W[k + koff];
        b.y = W[k + koff + 1];
        acc = __builtin_amdgcn_wmma_f32_16x16x4_f32(
            /*neg_a=*/false, a, /*neg_b=*/false, b,
            /*c_mod=*/(short)0, acc, /*reuse_a=*/false, /*reuse_b=*/false);
    }

    // C 16x16 layout: VGPR r -> (M = r + 8*hi, N = n0 + m)
    #pragma unroll
    for (int r = 0; r < 8; ++r) {
        int mm = r + hi * 8;
        int nn = n0 + m;
        C[mm][nn] = acc[r] + bk[iter * DIM + nn];
    }
    __syncthreads();

    // row L2 norms (x / max(||x||, 1e-12))
    if (t < 16) {
        float s = 0.0f;
        for (int d = 0; d < DIM; ++d) { float v = C[t][d]; s += v * v; }
        nrm[t] = fmaxf(sqrtf(s), 1e-12f);
    }
    __syncthreads();

    for (int i = t; i < 16 * DIM; i += 128) {
        int r = i >> 6, c = i & 63;
        emb[(size_t)(g0 + r) * DIM + c] = C[r][c] / nrm[r];
    }
}

// ---------------------------------------------------------------------------
// Final scoring: out[b] = sigmoid( relu((ge*ie) @ p1w^T + p1b) @ p2w^T + p2b )
// ---------------------------------------------------------------------------
__global__ void score_k(const int* __restrict__ groups,
                        const int* __restrict__ items,
                        const float* __restrict__ emb,
                        const float* __restrict__ item_emb,
                        const float* __restrict__ p1w,   // [16][64]
                        const float* __restrict__ p1b,   // [16]
                        const float* __restrict__ p2w,   // [1][16]
                        const float* __restrict__ p2b,   // [1]
                        float* __restrict__ out, int B) {
    int b = blockIdx.x * blockDim.x + threadIdx.x;
    if (b >= B) return;
    const float* ge = emb      + (size_t)groups[b] * DIM;
    const float* ie = item_emb + (size_t)items[b]  * DIM;
    float x[DIM];
    #pragma unroll
    for (int d = 0; d < DIM; ++d) x[d] = ge[d] * ie[d];
    float acc = p2b[0];
    #pragma unroll
    for (int j = 0; j < 16; ++j) {
        float h = p1b[j];
        const float* w = p1w + j * DIM;
        #pragma unroll
        for (int d = 0; d < DIM; ++d) h = fmaf(x[d], w[d], h);
        h = fmaxf(h, 0.0f);
        acc = fmaf(h, p2w[j], acc);
    }
    out[b] = 1.0f / (1.0f + expf(-acc));
}

// ---------------------------------------------------------------------------
extern "C" void kernel_launch(void* const* d_in, const int* in_sizes, int n_in,
                              void* d_out, int out_size, void* d_ws, size_t ws_size,
                              hipStream_t stream) {
    const int*   groups       = (const int*)  d_in[0];
    const int*   items        = (const int*)  d_in[1];
    const int*   membership   = (const int*)  d_in[2];
    const float* member_mask  = (const float*)d_in[3];
    /* d_in[4] hyper_graph: unused — factored out via membership (H has <=20 nnz/row) */
    const float* user_emb     = (const float*)d_in[5];
    const float* item_emb     = (const float*)d_in[6];
    const float* Wk           = (const float*)d_in[7];
    const float* bk           = (const float*)d_in[8];
    const float* p1w          = (const float*)d_in[9];
    const float* p1b          = (const float*)d_in[10];
    const float* p2w          = (const float*)d_in[11];
    const float* p2b          = (const float*)d_in[12];
    float*       out          = (float*)d_out;

    const int B  = in_sizes[0];
    const int G  = in_sizes[2] / MAXM;              // 2000
    const int NU = in_sizes[5] / DIM;               // 20000
    const int K  = in_sizes[7] / (DIM * 2 * DIM);   // 2

    // workspace layout (floats)
    float* ws           = (float*)d_ws;
    float* emb          = ws;                                   // G*64
    float* emb_cat      = emb          + (size_t)G * DIM;       // G*128
    float* neigh_member = emb_cat      + (size_t)G * 2 * DIM;   // G*64
    float* m_cnt        = neigh_member + (size_t)G * DIM;       // G
    float* deg          = m_cnt        + G;                     // NU
    float* user_acc     = deg          + NU;                    // NU*64

    zero_f32<<<(NU + 255) / 256, 256, 0, stream>>>(deg, NU);
    group_init_k<<<G, DIM, 0, stream>>>(membership, member_mask, user_emb,
                                        emb, m_cnt, deg);
    neigh_member_k<<<G, DIM, 0, stream>>>(membership, deg, user_emb, neigh_member);

    for (int it = 0; it < K; ++it) {
        zero_f32<<<(NU * DIM + 255) / 256, 256, 0, stream>>>(user_acc, NU * DIM);
        scatter_k<<<G, DIM, 0, stream>>>(membership, emb, user_acc);
        gather_cat_k<<<G, DIM, 0, stream>>>(membership, user_acc, emb,
                                            neigh_member, m_cnt, emb_cat);
        gcn_transform_wmma<<<G / 16, 128, 0, stream>>>(emb_cat, Wk, bk, emb, it);
    }

    score_k<<<(B + 127) / 128, 128, 0, stream>>>(groups, items, emb, item_emb,
                                                 p1w, p1b, p2w, p2b, out, B);
}